// AttentionPyramid_68779606278585
// MI455X (gfx1250) — compile-verified
//
#include <hip/hip_runtime.h>
#include <hip/hip_bf16.h>
#include <stdint.h>

// ---------------------------------------------------------------------------
// CDNA5 (gfx1250) WMMA fp32 helpers.
// V_WMMA_F32_16X16X4_F32:  D[16x16] = A[16x4] * B[4x16] + C
//   A: lane 0-15 hold rows M=0..15 with K={k0,k0+1}; lanes 16-31 K={k0+2,k0+3}
//   B: lane 0-15 hold cols N=0..15 with K={k0,k0+1}; lanes 16-31 K={k0+2,k0+3}
//   C/D: VGPR v holds row M = v + 8*(lane>=16), col N = lane&15
// ---------------------------------------------------------------------------
typedef __attribute__((ext_vector_type(2))) float v2f;
typedef __attribute__((ext_vector_type(8))) float v8f;

__device__ __forceinline__ v8f wmma_f32(v2f a, v2f b, v8f c) {
  return __builtin_amdgcn_wmma_f32_16x16x4_f32(false, a, false, b, (short)0, c,
                                               false, false);
}

__device__ __forceinline__ v2f frag_ld(const float* p, int kstride) {
  v2f r;
  r.x = p[0];
  r.y = p[kstride];
  return r;
}

__device__ __forceinline__ v8f v8zero() {
  v8f z;
#pragma unroll
  for (int i = 0; i < 8; ++i) z[i] = 0.0f;
  return z;
}

// Async DMA of 16 bytes/lane from global memory into LDS (gfx1250 VGLOBAL
// GLOBAL_LOAD_ASYNC_TO_LDS_B128, tracked by ASYNCcnt). ldsOff is the per-lane
// LDS byte address (low 32 bits of a generic pointer to __shared__ data).
__device__ __forceinline__ void async_ld_b128(uint32_t ldsOff, const float* g) {
  asm volatile("global_load_async_to_lds_b128 %0, %1, off"
               :
               : "v"(ldsOff), "v"(g)
               : "memory");
}
__device__ __forceinline__ void wait_async0() {
  asm volatile("s_wait_asynccnt 0" ::: "memory");
}

#define NSHOT 5
#define FSC 256

// ---------------------------------------------------------------------------
// 1x1 conv as GEMM: Y[nb][chOff+co][hw] = sum_ci W[co][ci] * X[nb][ci][hw] + b
// grid = (256/16, ceil(HW/16), NB), block = 32 (one wave per 16x16 tile)
// ---------------------------------------------------------------------------
__global__ __launch_bounds__(32) void conv1x1_wmma_k(
    const float* __restrict__ X, long inBatch, const float* __restrict__ W,
    const float* __restrict__ bias, float* __restrict__ Y, long outBatch,
    int outChOff, int Cin, int HW) {
  const int lane = threadIdx.x;
  const int half = lane >> 4, idx = lane & 15;
  const int co0 = blockIdx.x * 16;
  const int hw0 = blockIdx.y * 16;
  const int nb = blockIdx.z;
  const float* Xb = X + (long)nb * inBatch;
  const int n = hw0 + idx;
  const int nc = (n < HW) ? n : (HW - 1);
  const int kk = 2 * half;
  v8f acc = v8zero();
  for (int k0 = 0; k0 < Cin; k0 += 4) {
    v2f a = frag_ld(W + (long)(co0 + idx) * Cin + k0 + kk, 1);
    v2f b = frag_ld(Xb + (long)(k0 + kk) * HW + nc, HW);
    acc = wmma_f32(a, b, acc);
  }
  if (n < HW) {
    float* Yb = Y + (long)nb * outBatch + (long)outChOff * HW;
#pragma unroll
    for (int v = 0; v < 8; ++v) {
      const int m = co0 + v + 8 * half;
      Yb[(long)m * HW + n] = acc[v] + bias[m];
    }
  }
}

// ---------------------------------------------------------------------------
// 3x3 conv (pad=1, stride 1 or 2) as implicit GEMM over K = Cin*9.
// Optional ReLU on the input (for P7 = conv(relu(P6))).
// ---------------------------------------------------------------------------
__global__ __launch_bounds__(32) void conv3x3_wmma_k(
    const float* __restrict__ X, long inBatch, int Cin, int inH, int inW,
    const float* __restrict__ W, const float* __restrict__ bias,
    float* __restrict__ Y, long outBatch, int outChOff, int outH, int outW,
    int stride, int reluIn) {
  const int lane = threadIdx.x;
  const int half = lane >> 4, idx = lane & 15;
  const int co0 = blockIdx.x * 16;
  const int hw0 = blockIdx.y * 16;
  const int nb = blockIdx.z;
  const int HWo = outH * outW;
  const int inHW = inH * inW;
  const int n = hw0 + idx;
  const int np = (n < HWo) ? n : (HWo - 1);
  const int oy = np / outW, ox = np % outW;
  const int kk = 2 * half;
  const float* Xb = X + (long)nb * inBatch;
  v8f acc = v8zero();
  for (int tap = 0; tap < 9; ++tap) {
    const int iy = oy * stride + (tap / 3) - 1;
    const int ix = ox * stride + (tap % 3) - 1;
    const bool ok = (iy >= 0) && (iy < inH) && (ix >= 0) && (ix < inW);
    const long boff = ok ? ((long)iy * inW + ix) : 0;
    const float* Wt = W + (long)(co0 + idx) * Cin * 9 + tap;  // k-stride = 9
    for (int k0 = 0; k0 < Cin; k0 += 4) {
      v2f a = frag_ld(Wt + (long)(k0 + kk) * 9, 9);
      v2f b = frag_ld(Xb + (long)(k0 + kk) * inHW + boff, inHW);
      if (!ok) { b.x = 0.0f; b.y = 0.0f; }
      if (reluIn) { b.x = fmaxf(b.x, 0.0f); b.y = fmaxf(b.y, 0.0f); }
      acc = wmma_f32(a, b, acc);
    }
  }
  if (n < HWo) {
    float* Yb = Y + (long)nb * outBatch + (long)outChOff * HWo;
#pragma unroll
    for (int v = 0; v < 8; ++v) {
      const int m = co0 + v + 8 * half;
      Yb[(long)m * HWo + n] = acc[v] + bias[m];
    }
  }
}

// ---------------------------------------------------------------------------
// Linear layer: Y[bz][tok][n] = sum_k X(tok,k) * W[n][k] + bias[n]
// X element (tok,k) at X + bz*batchStride + tok*tokStride + k*kStride,
// so the same kernel reads token-major (K-proj) or channel-major (Q-proj from
// the NCHW feature map living in d_out).
// grid = (ceil(L/16), 16, NBatch)
// ---------------------------------------------------------------------------
__global__ __launch_bounds__(32) void linear_wmma_k(
    const float* __restrict__ X, long batchStride, long tokStride, long kStride,
    const float* __restrict__ W, const float* __restrict__ bias,
    float* __restrict__ Y, int L) {
  const int lane = threadIdx.x;
  const int half = lane >> 4, idx = lane & 15;
  const int mt = blockIdx.x, nt = blockIdx.y, bz = blockIdx.z;
  const int tok = mt * 16 + idx;
  const long tokc = (tok < L) ? tok : (L - 1);
  const float* Xb = X + (long)bz * batchStride;
  const int kk = 2 * half;
  v8f acc = v8zero();
  for (int k0 = 0; k0 < FSC; k0 += 4) {
    v2f a = frag_ld(Xb + tokc * tokStride + (long)(k0 + kk) * kStride,
                    (int)kStride);
    v2f b = frag_ld(W + (long)(nt * 16 + idx) * FSC + k0 + kk, 1);
    acc = wmma_f32(a, b, acc);
  }
  const int n = nt * 16 + idx;
  const float bv = bias[n];
  float* Yb = Y + (long)bz * L * FSC;
#pragma unroll
  for (int v = 0; v < 8; ++v) {
    const int t = mt * 16 + v + 8 * half;
    if (t < L) Yb[(long)t * FSC + n] = acc[v] + bv;
  }
}

// ---------------------------------------------------------------------------
// P4x += up2(P5x) style nearest 2x upsample-add (in place on dst).
// ---------------------------------------------------------------------------
__global__ void up2_add_k(float* __restrict__ dst, const float* __restrict__ src,
                          int NB, int H, int W) {
  const long i = (long)blockIdx.x * blockDim.x + threadIdx.x;
  const long total = (long)NB * FSC * H * W;
  if (i >= total) return;
  const int x = (int)(i % W);
  const int y = (int)((i / W) % H);
  const long cn = i / ((long)W * H);  // nb*256 + c
  dst[i] += src[cn * (long)(H / 2) * (W / 2) + (long)(y >> 1) * (W / 2) +
                (x >> 1)];
}

// ---------------------------------------------------------------------------
// V[s][b][l][c] = P[(b*5+s)][c][l] + PE(l,c)  (support split + transpose + PE)
// ---------------------------------------------------------------------------
__global__ void add_pe_k(const float* __restrict__ P, float* __restrict__ Vv,
                         int B, int L) {
  const long i = (long)blockIdx.x * blockDim.x + threadIdx.x;
  const long total = (long)NSHOT * B * L * FSC;
  if (i >= total) return;
  const int c = (int)(i & (FSC - 1));
  const int l = (int)((i >> 8) % L);
  const long sb = i / ((long)L * FSC);
  const int b = (int)(sb % B);
  const int s = (int)(sb / B);
  const float div = __expf(-(float)(c & ~1) * (9.210340371976184f / 256.0f));
  const float ang = (float)l * div;
  const float pe = (c & 1) ? __cosf(ang) : __sinf(ang);
  Vv[i] = P[((long)(b * NSHOT + s) * FSC + c) * L + l] + pe;
}

// ---------------------------------------------------------------------------
// Fused flash attention for one level.
//   out[b][256+c][l] = sum_s softmax_m(Qp[b,l]·Kp[s,b,m] / 16) @ V[s,b,m,c] / 5
// One wave per (16-query tile, b). K/V tiles are DMA'd into LDS with
// global_load_async_to_lds_b128 (ASYNCcnt), WMMAs are fed from LDS, online
// softmax over key tiles, per-shot normalized result accumulated into an LDS
// tile (deterministic, no atomics).
// ---------------------------------------------------------------------------
__global__ __launch_bounds__(32) void attn_flash_k(
    const float* __restrict__ Qp, const float* __restrict__ Kp,
    const float* __restrict__ V, float* __restrict__ Out, int B, int L) {
  __shared__ float Qs[16 * FSC];
  __shared__ float Ks[16 * FSC];
  __shared__ float Vs[16 * FSC];
  __shared__ float Os[16 * FSC];
  __shared__ float Ps[16 * 16];
  const int lane = threadIdx.x;
  const int half = lane >> 4, idx = lane & 15;
  const int lt = blockIdx.x, b = blockIdx.y;
  const int l0 = lt * 16;
  const int kk = 2 * half;

  for (int t = lane; t < 16 * FSC; t += 32) {
    const int r = t >> 8, c = t & (FSC - 1);
    const int tok = l0 + r;
    Qs[t] = (tok < L) ? Qp[((long)b * L + tok) * FSC + c] : 0.0f;
    Os[t] = 0.0f;
  }
  __syncthreads();

  const float scale = 0.0625f;  // 1/sqrt(256)
  for (int s = 0; s < NSHOT; ++s) {
    const float* Kb = Kp + (long)(s * B + b) * L * FSC;
    const float* Vb = V + (long)(s * B + b) * L * FSC;
    float mrow[8], lrow[8];
#pragma unroll
    for (int v = 0; v < 8; ++v) { mrow[v] = -3.0e38f; lrow[v] = 0.0f; }
    v8f o[16];
#pragma unroll
    for (int t = 0; t < 16; ++t) o[t] = v8zero();

    for (int kt = 0; kt < L; kt += 16) {
      // ---- async-DMA the 16x256 K and V tiles into LDS ------------------
      // 32 issues per tile: each moves 32 lanes x 16B = one half key row.
      for (int i = 0; i < 32; ++i) {
        const int fe = i * 128 + lane * 4;      // float index in tile
        const int r = fe >> 8, c = fe & (FSC - 1);
        const long gr = (kt + r < L) ? (long)(kt + r) : (long)(L - 1);
        async_ld_b128((uint32_t)(uintptr_t)&Ks[r * FSC + c],
                      Kb + gr * FSC + c);
        async_ld_b128((uint32_t)(uintptr_t)&Vs[r * FSC + c],
                      Vb + gr * FSC + c);
      }
      wait_async0();
      __syncthreads();

      // ---- S = Qp_tile @ Kp_tile^T  (K = 256), all operands in LDS ------
      v8f st = v8zero();
      for (int k0 = 0; k0 < FSC; k0 += 4) {
        v2f a = frag_ld(&Qs[idx * FSC + k0 + kk], 1);
        v2f kb = frag_ld(&Ks[idx * FSC + k0 + kk], 1);
        st = wmma_f32(a, kb, st);
      }
      const int key = kt + idx;
      const bool colOk = key < L;
      // online softmax: rows live in (v, half); columns striped on lanes 0..15
#pragma unroll
      for (int v = 0; v < 8; ++v) {
        float sv = colOk ? st[v] * scale : -3.0e38f;
        float mx = sv;
#pragma unroll
        for (int off = 8; off; off >>= 1)
          mx = fmaxf(mx, __shfl_xor(mx, off, 16));
        const float mnew = fmaxf(mrow[v], mx);
        float p = colOk ? __expf(sv - mnew) : 0.0f;
        float psum = p;
#pragma unroll
        for (int off = 8; off; off >>= 1) psum += __shfl_xor(psum, off, 16);
        const float corr = __expf(mrow[v] - mnew);
        lrow[v] = lrow[v] * corr + psum;
        mrow[v] = mnew;
#pragma unroll
        for (int t = 0; t < 16; ++t) o[t][v] *= corr;
        st[v] = p;
      }
      // reshape P (D layout) -> A layout through 1KB of LDS
      __syncthreads();
#pragma unroll
      for (int v = 0; v < 8; ++v) Ps[(v + 8 * half) * 16 + idx] = st[v];
      __syncthreads();
      // ---- O += P @ V_tile  (K = 16 keys, N = 256 channels), V in LDS ---
#pragma unroll
      for (int k0 = 0; k0 < 16; k0 += 4) {
        const int kr = k0 + kk;  // key row within tile (clamped rows have P=0)
        v2f a = frag_ld(&Ps[idx * 16 + k0 + kk], 1);
#pragma unroll
        for (int t = 0; t < 16; ++t) {
          v2f vb;
          vb.x = Vs[kr * FSC + t * 16 + idx];
          vb.y = Vs[(kr + 1) * FSC + t * 16 + idx];
          o[t] = wmma_f32(a, vb, o[t]);
        }
      }
      // drain LDS reads before the DMA overwrites Ks/Vs next iteration
      __syncthreads();
    }
    // fold this shot's normalized output into the LDS accumulator
#pragma unroll
    for (int t = 0; t < 16; ++t) {
#pragma unroll
      for (int v = 0; v < 8; ++v) {
        Os[(v + 8 * half) * FSC + t * 16 + idx] +=
            o[t][v] / (lrow[v] * (float)NSHOT);
      }
    }
  }
  // write the attended map into channels 256..511 (NCHW)
#pragma unroll
  for (int t = 0; t < 16; ++t) {
#pragma unroll
    for (int v = 0; v < 8; ++v) {
      const int tok = l0 + v + 8 * half;
      const int c = t * 16 + idx;
      if (tok < L)
        Out[((long)b * 512 + 256 + c) * L + tok] =
            Os[(v + 8 * half) * FSC + c];
    }
  }
}

// ---------------------------------------------------------------------------
// Orchestration
// ---------------------------------------------------------------------------
static inline dim3 grid1d(long n) { return dim3((unsigned)((n + 255) / 256)); }

extern "C" void kernel_launch(void* const* d_in, const int* in_sizes, int n_in,
                              void* d_out, int out_size, void* d_ws,
                              size_t ws_size, hipStream_t stream) {
  (void)in_sizes; (void)n_in; (void)out_size; (void)ws_size;
  const float* C3 = (const float*)d_in[0];
  const float* C4 = (const float*)d_in[1];
  const float* C5 = (const float*)d_in[2];
  const float* S3 = (const float*)d_in[3];
  const float* S4 = (const float*)d_in[4];
  const float* S5 = (const float*)d_in[5];
  const float* w_p3_1 = (const float*)d_in[6];
  const float* b_p3_1 = (const float*)d_in[7];
  const float* w_p3_2 = (const float*)d_in[8];
  const float* b_p3_2 = (const float*)d_in[9];
  const float* w_p4_1 = (const float*)d_in[10];
  const float* b_p4_1 = (const float*)d_in[11];
  const float* w_p4_2 = (const float*)d_in[12];
  const float* b_p4_2 = (const float*)d_in[13];
  const float* w_p5_1 = (const float*)d_in[14];
  const float* b_p5_1 = (const float*)d_in[15];
  const float* w_p5_2 = (const float*)d_in[16];
  const float* b_p5_2 = (const float*)d_in[17];
  const float* w_p6 = (const float*)d_in[18];
  const float* b_p6 = (const float*)d_in[19];
  const float* w_p7 = (const float*)d_in[20];
  const float* b_p7 = (const float*)d_in[21];
  const float* wq = (const float*)d_in[22];
  const float* bq = (const float*)d_in[23];
  const float* wk = (const float*)d_in[24];
  const float* bk = (const float*)d_in[25];

  float* out = (float*)d_out;
  float* ws = (float*)d_ws;

  // workspace layout (floats); PSUP doubles as the K-projection buffer
  const size_t OFF_P5X = 0;                               //  8*256*100
  const size_t OFF_P4X = OFF_P5X + (size_t)8 * 256 * 100; //  8*256*400
  const size_t OFF_P3X = OFF_P4X + (size_t)8 * 256 * 400; //  8*256*1600
  const size_t OFF_P6S = OFF_P3X + (size_t)8 * 256 * 1600;    // 40*256*25
  const size_t OFF_PSUP = OFF_P6S + (size_t)40 * 256 * 25;    // 40*256*1600
  const size_t OFF_V = OFF_PSUP + (size_t)40 * 256 * 1600;    // 5*8*1600*256
  const size_t OFF_QP = OFF_V + (size_t)5 * 8 * 1600 * 256;   // 8*1600*256

  const size_t DO[5] = {0, 6553600, 8192000, 8601600, 8704000};
  const int Ls[5] = {1600, 400, 100, 25, 9};
  const dim3 W32(32);

  // ---- query FPN ---------------------------------------------------------
  conv1x1_wmma_k<<<dim3(16, 7, 8), W32, 0, stream>>>(
      C5, (long)2048 * 100, w_p5_1, b_p5_1, ws + OFF_P5X, (long)256 * 100, 0,
      2048, 100);
  conv1x1_wmma_k<<<dim3(16, 25, 8), W32, 0, stream>>>(
      C4, (long)1024 * 400, w_p4_1, b_p4_1, ws + OFF_P4X, (long)256 * 400, 0,
      1024, 400);
  conv1x1_wmma_k<<<dim3(16, 100, 8), W32, 0, stream>>>(
      C3, (long)512 * 1600, w_p3_1, b_p3_1, ws + OFF_P3X, (long)256 * 1600, 0,
      512, 1600);
  up2_add_k<<<grid1d((long)8 * 256 * 400), 256, 0, stream>>>(
      ws + OFF_P4X, ws + OFF_P5X, 8, 20, 20);
  up2_add_k<<<grid1d((long)8 * 256 * 1600), 256, 0, stream>>>(
      ws + OFF_P3X, ws + OFF_P4X, 8, 40, 40);
  // smoothing heads straight into d_out channels [0,256)
  conv3x3_wmma_k<<<dim3(16, 7, 8), W32, 0, stream>>>(
      ws + OFF_P5X, (long)256 * 100, 256, 10, 10, w_p5_2, b_p5_2, out + DO[2],
      (long)512 * 100, 0, 10, 10, 1, 0);
  conv3x3_wmma_k<<<dim3(16, 25, 8), W32, 0, stream>>>(
      ws + OFF_P4X, (long)256 * 400, 256, 20, 20, w_p4_2, b_p4_2, out + DO[1],
      (long)512 * 400, 0, 20, 20, 1, 0);
  conv3x3_wmma_k<<<dim3(16, 100, 8), W32, 0, stream>>>(
      ws + OFF_P3X, (long)256 * 1600, 256, 40, 40, w_p3_2, b_p3_2, out + DO[0],
      (long)512 * 1600, 0, 40, 40, 1, 0);
  conv3x3_wmma_k<<<dim3(16, 2, 8), W32, 0, stream>>>(
      C5, (long)2048 * 100, 2048, 10, 10, w_p6, b_p6, out + DO[3],
      (long)512 * 25, 0, 5, 5, 2, 0);
  conv3x3_wmma_k<<<dim3(16, 1, 8), W32, 0, stream>>>(
      out + DO[3], (long)512 * 25, 256, 5, 5, w_p7, b_p7, out + DO[4],
      (long)512 * 9, 0, 3, 3, 2, 1);

  // ---- per-level support features + attention ----------------------------
  for (int lvl = 0; lvl < 5; ++lvl) {
    const int L = Ls[lvl];
    const int lt = (L + 15) / 16;
    const float* supFeat = ws + OFF_PSUP;
    if (lvl == 0) {
      conv1x1_wmma_k<<<dim3(16, 100, 40), W32, 0, stream>>>(
          S3, (long)512 * 1600, w_p3_1, b_p3_1, ws + OFF_PSUP,
          (long)256 * 1600, 0, 512, 1600);
    } else if (lvl == 1) {
      conv1x1_wmma_k<<<dim3(16, 25, 40), W32, 0, stream>>>(
          S4, (long)1024 * 400, w_p4_1, b_p4_1, ws + OFF_PSUP, (long)256 * 400,
          0, 1024, 400);
    } else if (lvl == 2) {
      conv1x1_wmma_k<<<dim3(16, 7, 40), W32, 0, stream>>>(
          S5, (long)2048 * 100, w_p5_1, b_p5_1, ws + OFF_PSUP, (long)256 * 100,
          0, 2048, 100);
    } else if (lvl == 3) {
      // P6s lives in its own region so lvl-4's P7s conv can still read it
      conv3x3_wmma_k<<<dim3(16, 2, 40), W32, 0, stream>>>(
          S5, (long)2048 * 100, 2048, 10, 10, w_p6, b_p6, ws + OFF_P6S,
          (long)256 * 25, 0, 5, 5, 2, 0);
      supFeat = ws + OFF_P6S;
    } else {
      conv3x3_wmma_k<<<dim3(16, 1, 40), W32, 0, stream>>>(
          ws + OFF_P6S, (long)256 * 25, 256, 5, 5, w_p7, b_p7, ws + OFF_PSUP,
          (long)256 * 9, 0, 3, 3, 2, 1);
    }
    // V[s][b][l][c] = support + positional encoding
    add_pe_k<<<grid1d((long)NSHOT * 8 * L * 256), 256, 0, stream>>>(
        supFeat, ws + OFF_V, 8, L);
    // Q projection from NCHW query map in d_out (channel-major reads)
    linear_wmma_k<<<dim3(lt, 16, 8), W32, 0, stream>>>(
        out + DO[lvl], (long)512 * L, (long)1, (long)L, wq, bq, ws + OFF_QP, L);
    // K projection from token-major V (PSUP region is dead -> reuse for Kp)
    linear_wmma_k<<<dim3(lt, 16, 40), W32, 0, stream>>>(
        ws + OFF_V, (long)L * 256, (long)256, (long)1, wk, bk, ws + OFF_PSUP,
        L);
    attn_flash_k<<<dim3(lt, 8), W32, 0, stream>>>(
        ws + OFF_QP, ws + OFF_PSUP, ws + OFF_V, out + DO[lvl], 8, L);
  }
}